// GCN_34548716929331
// MI455X (gfx1250) — compile-verified
//
#include <hip/hip_runtime.h>

#define NNODES 100000
#define NEDGES 1600000
#define INDIM  128
#define HID    32

typedef __attribute__((ext_vector_type(2))) float v2f;
typedef __attribute__((ext_vector_type(8))) float v8f;

// ---------------- normalization ----------------

__global__ void k_deg_init(float* __restrict__ deg, int n) {
    int i = blockIdx.x * blockDim.x + threadIdx.x;
    if (i < n) deg[i] = 1.0f;   // self-loop contributes 1 to every node's degree
}

__global__ void k_deg_accum(const int* __restrict__ dst, float* __restrict__ deg, int e) {
    int i = blockIdx.x * blockDim.x + threadIdx.x;
    if (i < e) atomicAdd(&deg[dst[i]], 1.0f);
}

__global__ void k_rsqrt(float* __restrict__ deg, int n) {
    int i = blockIdx.x * blockDim.x + threadIdx.x;
    if (i < n) {
        float d = deg[i];
        deg[i] = (d > 0.0f) ? rsqrtf(d) : 0.0f;
    }
}

__global__ void k_norm(const int* __restrict__ src, const int* __restrict__ dst,
                       const float* __restrict__ dis, float* __restrict__ norm, int e) {
    int i = blockIdx.x * blockDim.x + threadIdx.x;
    if (i < e) norm[i] = dis[src[i]] * dis[dst[i]];
}

// ---------------- dense transform via V_WMMA_F32_16X16X4_F32 ----------------
// Each wave computes a 16-row x 32-col tile of Y = X(nrows x K) * W(K x 32).
// A-matrix 16x4 f32 layout: lanes 0-15 hold row M=lane; VGPR0 = K { lane<16 ? 0 : 2 },
// VGPR1 = K+1.  B 4x16: lanes 0-15 N=lane; VGPR0 = K {0|2}, VGPR1 = K+1.
__global__ __launch_bounds__(256) void k_gemm_wmma(const float* __restrict__ X,
                                                   const float* __restrict__ W,
                                                   float* __restrict__ Y,
                                                   int nrows, int K) {
    const int lane    = threadIdx.x & 31;
    const int wave    = threadIdx.x >> 5;
    const int laneHi  = lane >> 4;        // 0 or 1 (selects K pair 0/1 vs 2/3)
    const int laneLo  = lane & 15;
    const int rowbase = (blockIdx.x * 8 + wave) * 16;
    if (rowbase >= nrows) return;         // wave-uniform: EXEC stays all-1s for WMMA

    int mrow = rowbase + laneLo;
    if (mrow > nrows - 1) mrow = nrows - 1;           // clamp loads (stores guarded)
    const float* __restrict__ xrow = X + (size_t)mrow * (size_t)K;

    v8f c0 = {};   // columns 0..15
    v8f c1 = {};   // columns 16..31
    for (int kb = 0; kb < K; kb += 4) {
        const int kq = kb + 2 * laneHi;
        v2f a;
        a.x = xrow[kq];
        a.y = xrow[kq + 1];
        v2f b0, b1;
        b0.x = W[(size_t)kq * HID + laneLo];
        b0.y = W[(size_t)(kq + 1) * HID + laneLo];
        b1.x = W[(size_t)kq * HID + 16 + laneLo];
        b1.y = W[(size_t)(kq + 1) * HID + 16 + laneLo];
        c0 = __builtin_amdgcn_wmma_f32_16x16x4_f32(false, a, false, b0, (short)0, c0, false, false);
        c1 = __builtin_amdgcn_wmma_f32_16x16x4_f32(false, a, false, b1, (short)0, c1, false, false);
    }

    // C/D 16x16 f32 layout: VGPR r -> (M = r + 8*laneHi, N = laneLo)
    #pragma unroll
    for (int r = 0; r < 8; ++r) {
        int row = rowbase + r + 8 * laneHi;
        if (row < nrows) {
            Y[(size_t)row * HID + laneLo]      = c0[r];
            Y[(size_t)row * HID + 16 + laneLo] = c1[r];
        }
    }
}

// ---------------- aggregation ----------------

// agg[i,f] = t[i,f] * dis[i]^2   (self-loop term; also initializes poisoned buffer)
__global__ void k_selfloop_init(const float* __restrict__ t, const float* __restrict__ dis,
                                float* __restrict__ agg, int n) {
    int i = blockIdx.x * blockDim.x + threadIdx.x;
    if (i < n * HID) {
        float d = dis[i >> 5];
        agg[i] = t[i] * d * d;
    }
}

// One lane per (edge, feature): coalesced gather load + coalesced f32 atomic add.
__global__ void k_scatter(const int* __restrict__ src, const int* __restrict__ dst,
                          const float* __restrict__ norm, const float* __restrict__ t,
                          float* __restrict__ agg, int e) {
    int i    = blockIdx.x * blockDim.x + threadIdx.x;
    int edge = i >> 5;
    int f    = i & 31;
    if (edge < e) {
        int   s = src[edge];
        int   d = dst[edge];
        float w = norm[edge];
        atomicAdd(&agg[(size_t)d * HID + f], t[(size_t)s * HID + f] * w);
    }
}

__global__ void k_bias_act(const float* __restrict__ agg, const float* __restrict__ b,
                           float* __restrict__ out, int n, int do_relu) {
    int i = blockIdx.x * blockDim.x + threadIdx.x;
    if (i < n * HID) {
        float v = agg[i] + b[i & 31];
        out[i] = do_relu ? fmaxf(v, 0.0f) : v;
    }
}

// ---------------- launch ----------------

extern "C" void kernel_launch(void* const* d_in, const int* in_sizes, int n_in,
                              void* d_out, int out_size, void* d_ws, size_t ws_size,
                              hipStream_t stream) {
    const float* x   = (const float*)d_in[0];
    const int*   ei  = (const int*)d_in[1];      // [2, E] int32 (JAX x64 disabled)
    const float* W0  = (const float*)d_in[2];
    const float* b0  = (const float*)d_in[3];
    const float* W1  = (const float*)d_in[4];
    const float* b1  = (const float*)d_in[5];
    const float* W2  = (const float*)d_in[6];
    const float* b2  = (const float*)d_in[7];

    const int* src = ei;            // edge_index[0]
    const int* dst = ei + NEDGES;   // edge_index[1]

    auto alignup = [](size_t v) { return (v + 255) & ~(size_t)255; };
    char*  ws   = (char*)d_ws;
    size_t off  = 0;
    float* dis  = (float*)(ws + off); off += alignup((size_t)NNODES * 4);
    float* nrm  = (float*)(ws + off); off += alignup((size_t)NEDGES * 4);
    float* t    = (float*)(ws + off); off += alignup((size_t)NNODES * HID * 4);
    float* h    = (float*)(ws + off); off += alignup((size_t)NNODES * HID * 4);
    float* out  = (float*)d_out;

    const int TB = 256;
    const int gN   = (NNODES + TB - 1) / TB;
    const int gE   = (NEDGES + TB - 1) / TB;
    const int gNF  = (NNODES * HID + TB - 1) / TB;
    const int gEF  = (int)(((size_t)NEDGES * HID + TB - 1) / TB);
    const int gGem = (NNODES + 127) / 128;       // 8 waves/block * 16 rows/wave

    // Symmetric GCN normalization (shared across layers)
    k_deg_init <<<gN, TB, 0, stream>>>(dis, NNODES);
    k_deg_accum<<<gE, TB, 0, stream>>>(dst, dis, NEDGES);
    k_rsqrt    <<<gN, TB, 0, stream>>>(dis, NNODES);
    k_norm     <<<gE, TB, 0, stream>>>(src, dst, dis, nrm, NEDGES);

    // Layer 0: t = x @ W0 ; h = scatter(t) ; h = relu(h + b0)
    k_gemm_wmma    <<<gGem, TB, 0, stream>>>(x, W0, t, NNODES, INDIM);
    k_selfloop_init<<<gNF,  TB, 0, stream>>>(t, dis, h, NNODES);
    k_scatter      <<<gEF,  TB, 0, stream>>>(src, dst, nrm, t, h, NEDGES);
    k_bias_act     <<<gNF,  TB, 0, stream>>>(h, b0, h, NNODES, 1);

    // Layer 1
    k_gemm_wmma    <<<gGem, TB, 0, stream>>>(h, W1, t, NNODES, HID);
    k_selfloop_init<<<gNF,  TB, 0, stream>>>(t, dis, h, NNODES);
    k_scatter      <<<gEF,  TB, 0, stream>>>(src, dst, nrm, t, h, NEDGES);
    k_bias_act     <<<gNF,  TB, 0, stream>>>(h, b1, h, NNODES, 1);

    // Layer 2 (no relu), aggregate directly into d_out
    k_gemm_wmma    <<<gGem, TB, 0, stream>>>(h, W2, t, NNODES, HID);
    k_selfloop_init<<<gNF,  TB, 0, stream>>>(t, dis, out, NNODES);
    k_scatter      <<<gEF,  TB, 0, stream>>>(src, dst, nrm, t, out, NEDGES);
    k_bias_act     <<<gNF,  TB, 0, stream>>>(out, b2, out, NNODES, 0);

    (void)in_sizes; (void)n_in; (void)out_size; (void)ws_size;
}